// TeacherEmbeddingTransform_51539607702
// MI455X (gfx1250) — compile-verified
//
#include <hip/hip_runtime.h>

// Block-diagonal GEMM: out[16384,576] = blockdiag(x[16384,2304] @ W) + bias
// 64 blocks: 16x(K=16,N=4), 32x(K=32,N=8), 16x(K=64,N=16).
// HBM-bound (~4.4 FLOP/B -> ~8us floor at 23.3 TB/s); f32 end-to-end, so the
// matrix op is V_WMMA_F32_16X16X4_F32. Geometry is compile-time per block
// type (scalar blockIdx.y select) so loops fully unroll with no exec masking.
// Weights are pre-packed into d_ws in WMMA B-fragment order (zero-padded to
// N=16) so the hot loop's B fetch is a single coalesced b64 load per lane.

typedef float v2f __attribute__((ext_vector_type(2)));
typedef float v8f __attribute__((ext_vector_type(8)));

#define IN_DIM       2304
#define OUT_DIM      576
#define NTILES       1024        // 16384 batch rows / 16 rows per tile
#define WAVES_PER_WG 8
#define PACK_FLOATS  36864       // 16 * sum(SIZE) = 16*2304 packed B floats

// ---------------------------------------------------------------------------
// Prep: pack W (row-major size x choice per block) into WMMA B-fragment order.
// Layout per block: step s (K=4s..4s+3) -> 32 lanes x float2 (64 floats).
// Lane l: half=l>>4, n=l&15;  pv = { W[4s+2*half][n], W[4s+2*half+1][n] },
// zero for n >= choice. One workgroup per logical block.
// ---------------------------------------------------------------------------
__global__ __launch_bounds__(256)
void pack_w_kernel(const float* __restrict__ w, float* __restrict__ wpak) {
    const int blk = blockIdx.x;                 // 0..63 (scalar)
    int size, choice, w_off, pak_off;
    if (blk < 16)      { size = 16; choice = 4;
                         w_off = 64 * blk;             pak_off = 256 * blk; }
    else if (blk < 48) { const int j = blk - 16; size = 32; choice = 8;
                         w_off = 1024 + 256 * j;       pak_off = 4096 + 512 * j; }
    else               { const int j = blk - 48; size = 64; choice = 16;
                         w_off = 9216 + 1024 * j;      pak_off = 20480 + 1024 * j; }

    const int lane = threadIdx.x & 31;
    const int wv   = threadIdx.x >> 5;          // 0..7
    const int half = lane >> 4;
    const int n    = lane & 15;
    const int nsteps = size >> 2;

    for (int s = wv; s < nsteps; s += WAVES_PER_WG) {
        const int ka = 4 * s + 2 * half;
        v2f pv; pv.x = 0.f; pv.y = 0.f;
        if (n < choice) {
            pv.x = w[w_off + ka * choice + n];
            pv.y = w[w_off + (ka + 1) * choice + n];
        }
        *(v2f*)(wpak + pak_off + s * 64 + lane * 2) = pv;
    }
}

// ---------------------------------------------------------------------------
// Main compute body: one wave = one 16-row batch tile x one block.
// Fully unrolled SIZE/4 steps of V_WMMA_F32_16X16X4_F32, two accumulators.
// ---------------------------------------------------------------------------
template <int SIZE, int CHOICE, bool PACKED>
__device__ __forceinline__
void bdiag_body(const float* __restrict__ x, const float* __restrict__ wsrc,
                const float* __restrict__ bias, float* __restrict__ out,
                int b, int in_base, int out_base, int w_base) {
    const int lane = threadIdx.x & 31;
    const int half = lane >> 4;                 // K-pair selector
    const int n    = lane & 15;                 // A: row M; B/D: col N
    const int tile = blockIdx.x * WAVES_PER_WG + (threadIdx.x >> 5);

    const int in_off  = in_base  + SIZE   * b;
    const int out_off = out_base + CHOICE * b;

    // A fragment base: row (tile*16 + n), this lane's K pair starts at 2*half
    const float* __restrict__ xrow =
        x + (size_t)(tile * 16 + n) * IN_DIM + in_off + 2 * half;

    // B fragment base
    const float* __restrict__ bptr;
    if (PACKED) bptr = wsrc + w_base + 16 * SIZE * b + lane * 2;
    else        bptr = wsrc + w_base + SIZE * CHOICE * b + 2 * half * CHOICE + n;

    constexpr int NSTEP = SIZE / 4;
    v8f acc0 = {0.f, 0.f, 0.f, 0.f, 0.f, 0.f, 0.f, 0.f};
    v8f acc1 = {0.f, 0.f, 0.f, 0.f, 0.f, 0.f, 0.f, 0.f};

#pragma unroll
    for (int s = 0; s < NSTEP; ++s) {
        v2f a = *(const v2f*)(xrow + 4 * s);            // global_load_b64

        v2f bf;
        if (PACKED) {
            bf = *(const v2f*)(bptr + 64 * s);          // coalesced b64
        } else {
            if (CHOICE == 16 || n < CHOICE) {
                bf.x = bptr[(4 * s) * CHOICE];
                bf.y = bptr[(4 * s + 1) * CHOICE];
            } else { bf.x = 0.f; bf.y = 0.f; }
        }

        if (s & 1)
            acc1 = __builtin_amdgcn_wmma_f32_16x16x4_f32(
                false, a, false, bf, (short)0, acc1, false, false);
        else
            acc0 = __builtin_amdgcn_wmma_f32_16x16x4_f32(
                false, a, false, bf, (short)0, acc0, false, false);
    }

    // D layout: VGPR i -> M = i + 8*half, N = n
    if (CHOICE == 16 || n < CHOICE) {
        const float bs = bias[out_off + n];
        float* __restrict__ op =
            out + (size_t)(tile * 16 + half * 8) * OUT_DIM + out_off + n;
#pragma unroll
        for (int i = 0; i < 8; ++i)
            op[(size_t)i * OUT_DIM] = acc0[i] + acc1[i] + bs;
    }
}

// One fused kernel: blockIdx.y (scalar) selects the compile-time block type.
template <bool PACKED>
__global__ __launch_bounds__(256)
void bdiag_gemm_kernel(const float* __restrict__ x,
                       const float* __restrict__ wsrc,
                       const float* __restrict__ bias,
                       float* __restrict__ out) {
    const int blk = blockIdx.y;                 // 0..63, uniform
    if (blk < 16) {
        bdiag_body<16, 4, PACKED>(x, wsrc, bias, out, blk,
                                  /*in*/0, /*out*/0,
                                  /*w*/PACKED ? 0 : 0);
    } else if (blk < 48) {
        bdiag_body<32, 8, PACKED>(x, wsrc, bias, out, blk - 16,
                                  /*in*/256, /*out*/64,
                                  /*w*/PACKED ? 4096 : 1024);
    } else {
        bdiag_body<64, 16, PACKED>(x, wsrc, bias, out, blk - 48,
                                   /*in*/1280, /*out*/320,
                                   /*w*/PACKED ? 20480 : 9216);
    }
}

extern "C" void kernel_launch(void* const* d_in, const int* in_sizes, int n_in,
                              void* d_out, int out_size, void* d_ws, size_t ws_size,
                              hipStream_t stream) {
    const float* x    = (const float*)d_in[0];   // [16384, 2304] f32
    const float* w    = (const float*)d_in[1];   // [25600, 1]    f32
    const float* bias = (const float*)d_in[2];   // [576]         f32
    float*       out  = (float*)d_out;           // [16384, 576]  f32

    const dim3 grid(NTILES / WAVES_PER_WG, 64);  // (128, 64) workgroups
    const dim3 block(256);

    if (ws_size >= (size_t)PACK_FLOATS * sizeof(float)) {
        float* wpak = (float*)d_ws;
        hipLaunchKernelGGL(pack_w_kernel, dim3(64), block, 0, stream, w, wpak);
        hipLaunchKernelGGL((bdiag_gemm_kernel<true>), grid, block, 0, stream,
                           x, wpak, bias, out);
    } else {
        hipLaunchKernelGGL((bdiag_gemm_kernel<false>), grid, block, 0, stream,
                           x, w, bias, out);
    }
}